// BidirectionalTrustModel_26396869001245
// MI455X (gfx1250) — compile-verified
//
#include <hip/hip_runtime.h>

// ---------------------------------------------------------------------------
// BidirectionalTrustModel on MI455X (gfx1250, wave32).
//
// Memory bound: 48 MB streamed once -> ~2us at 23.3 TB/s. No matmul structure,
// so WMMA does not apply; the CDNA5 paths used are:
//  * Tensor Data Mover: one tensor_load_to_lds per block DMAs the 2D strided
//    perf tile (64 rows x 64B, row stride B*8) into LDS (TENSORcnt).
//  * ASYNCcnt DMA: global_load_async_to_lds_b32 stages the ids slab.
//  * wave32-native layout: one wave per sequence, lane == channel (C==32).
//  * scan algebra: a fold of max/min-with-constant is x->min(max(x,lo),hi);
//    (lo,hi) pre-baked per (op,id,channel) in LDS, so each of the 64 steps is
//    one ds_load_b64 + v_max_num_f32 + v_min_num_f32, issued in groups of 8.
//  * per-step (op,id) broadcast via v_readlane (imm lane, fully unrolled).
//  * trust = (ballot_w32(required <= mean) == 0xFFFFFFFF).
// ---------------------------------------------------------------------------

typedef __attribute__((ext_vector_type(4))) unsigned u32x4_t;
typedef __attribute__((ext_vector_type(8))) unsigned u32x8_t;

#define FLT_MAX_C 3.402823466e+38f

#define TT      64   // tasks per sequence
#define CC      32   // capability channels == wave32 width
#define NTASK   6
#define BPB     8    // sequences (waves) per block
#define THREADS 256  // 8 wave32 waves

// Low 32 bits of a generic pointer to __shared__ are the LDS byte address
// (ISA 10.2).
__device__ __forceinline__ unsigned lds_addr32(const void* p) {
    return (unsigned)(unsigned long long)p;
}

__global__ __launch_bounds__(THREADS) void trust_scan_kernel(
    const unsigned long long* __restrict__ perf,   // [T*B] packed {p0(lo32), p1(hi32)}
    const int*                __restrict__ obsids, // [T*B]
    const int*                __restrict__ predids,// [B]
    const float*              __restrict__ obsM,   // [C*NTASK] row-major [c][j]
    float*                    __restrict__ out,    // [B]
    int Bn)
{
    __shared__ unsigned long long sPerf[TT][BPB];        // 4 KB, filled by TDM
    __shared__ int                sId[TT][BPB];          // 2 KB, filled by async DMA
    __shared__ float2             sTbl[3 * NTASK][CC];   // 4.5 KB (lo,hi) per enc,channel
    __shared__ int                sPid[BPB];

    const int  tid     = threadIdx.x;
    const long b0      = (long)blockIdx.x * BPB;
    const bool tdmWave = (tid < 32);                     // wave 0 drives the TDM

    // ---- TDM: DMA the perf tile Memory->LDS with one instruction.
    //      D# (ISA 8.3/8.4): 2D tile, data_size=8B, tile = 8 x 64,
    //      tensor_dim0 = tensor_dim0_stride = Bn elements.
    if (tdmWave) {
        unsigned long long ga = (unsigned long long)(const void*)(perf + b0);
        unsigned dim0 = (unsigned)Bn;
        u32x4_t g0, gz;
        u32x8_t g1;
        g0[0] = 1u;                                      // count=1 (valid), flags=0
        g0[1] = lds_addr32(&sPerf[0][0]);                // lds_addr
        g0[2] = (unsigned)ga;                            // global_addr[31:0]
        g0[3] = (unsigned)(ga >> 32) | (2u << 30);       // global_addr[56:32] | type=2
        g1[0] = 3u << 16;                                // wg_mask=0 | data_size=8B
        g1[1] = (dim0 & 0xFFFFu) << 16;                  // tensor_dim0[15:0] @ 63:48
        g1[2] = (dim0 >> 16) | ((unsigned)TT << 16);     // dim0[31:16] | tensor_dim1=64
        g1[3] = (unsigned)BPB << 16;                     // dim1[31:16]=0 | tile_dim0=8
        g1[4] = (unsigned)TT;                            // tile_dim1=64 | tile_dim2=0
        g1[5] = dim0;                                    // tensor_dim0_stride[31:0]
        g1[6] = 0u;                                      // stride0[47:32] | stride1[15:0]
        g1[7] = 0u;                                      // stride1[47:16]
        gz[0] = gz[1] = gz[2] = gz[3] = 0u;              // groups 2/3 unused (2D tile)
        asm volatile("tensor_load_to_lds %0, %1, %2, %3"
                     :: "s"(g0), "s"(g1), "s"(gz), "s"(gz) : "memory");
    }

    // ---- ids slab via the async-to-LDS path (512 b32, 2 per thread).
#pragma unroll
    for (int r = 0; r < 2; ++r) {
        int i  = tid + r * THREADS;                      // 0..511
        int t  = i >> 3, bl = i & 7;
        unsigned   dst4 = lds_addr32(&sId[t][bl]);
        const int* src4 = &obsids[(long)t * Bn + b0 + bl];
        asm volatile("global_load_async_to_lds_b32 %0, %1, off"
                     :: "v"(dst4), "v"(src4) : "memory");
    }

    // ---- Build the (op,id)->(lo,hi) clamp table while both DMAs fly.
    //      enc = op*6+id.  op0: (-MAX,+MAX)  op1(max): (M,+MAX)  op2(min): (-MAX,M)
    if (tid < NTASK * CC) {
        int   id = tid >> 5, c = tid & 31;
        float m  = obsM[c * NTASK + id];
        sTbl[id][c]             = make_float2(-FLT_MAX_C, FLT_MAX_C);
        sTbl[NTASK + id][c]     = make_float2(m,          FLT_MAX_C);
        sTbl[2 * NTASK + id][c] = make_float2(-FLT_MAX_C, m);
    }
    if (tid < BPB) sPid[tid] = predids[b0 + tid];

    if (tdmWave) __builtin_amdgcn_s_wait_tensorcnt(0);   // own TENSORcnt drained
    asm volatile("s_wait_asynccnt 0" ::: "memory");      // own ASYNCcnt drained
    __syncthreads();                                     // publish LDS to all waves

    const int lane = tid & 31;   // channel
    const int w    = tid >> 5;   // which of the 8 sequences this wave owns

    // ---- Lane t holds step t's code; decode to enc = op*6+id (per lane).
    unsigned long long pv0 = sPerf[lane][w];
    unsigned long long pv1 = sPerf[lane + 32][w];
    int id0 = sId[lane][w];
    int id1 = sId[lane + 32][w];

    int p0a  = ((unsigned)(pv0 & 0xffffffffull) != 0u);
    int p1a  = ((unsigned)(pv0 >> 32)           != 0u);
    int op0  = (!p0a && p1a) ? 1 : ((p0a && !p1a) ? 2 : 0);
    int enc0 = op0 * NTASK + id0;

    int p0b  = ((unsigned)(pv1 & 0xffffffffull) != 0u);
    int p1b  = ((unsigned)(pv1 >> 32)           != 0u);
    int op1  = (!p0b && p1b) ? 1 : ((p0b && !p1b) ? 2 : 0);
    int enc1 = op1 * NTASK + id1;

    // ---- Sequential scan, 8 steps per group: 8 ds_load_b64 in flight, then
    //      the (serial) v_max/v_min chain consumes them in order.
    float mean = 0.0f;
#pragma unroll
    for (int t0 = 0; t0 < TT; t0 += 8) {
        float2 p[8];
#pragma unroll
        for (int k = 0; k < 8; ++k) {
            const int t = t0 + k;
            int e = __builtin_amdgcn_readlane((t < 32) ? enc0 : enc1, t & 31); // uniform
            p[k]  = sTbl[e][lane];
        }
#pragma unroll
        for (int k = 0; k < 8; ++k)
            mean = fminf(fmaxf(mean, p[k].x), p[k].y);
    }

    // ---- trust[b] = all_c(required[c] <= mean[c]) via wave32 ballot.
    int   pid = sPid[w];
    float req = sTbl[NTASK + pid][lane].x;  // op==1 row stores M[c][pid] in .x
    bool  ok  = (req <= mean);
#if __has_builtin(__builtin_amdgcn_ballot_w32)
    unsigned mask = __builtin_amdgcn_ballot_w32(ok);
#else
    unsigned mask = (unsigned)__ballot(ok);
#endif
    if (lane == 0) out[b0 + w] = (mask == 0xFFFFFFFFu) ? 1.0f : 0.0f;
}

extern "C" void kernel_launch(void* const* d_in, const int* in_sizes, int n_in,
                              void* d_out, int out_size, void* d_ws, size_t ws_size,
                              hipStream_t stream)
{
    (void)n_in; (void)d_ws; (void)ws_size; (void)out_size;
    // setup_inputs() order: inptasksperf[T,B,2] i32, tasksobsids[T,B,1] i32,
    //                       taskspredids[B,1] i32, obsMatrix[C,6] f32
    const unsigned long long* perf    = (const unsigned long long*)d_in[0]; // int2 pairs
    const int*                obsids  = (const int*)d_in[1];
    const int*                predids = (const int*)d_in[2];
    const float*              obsM    = (const float*)d_in[3];
    float*                    out     = (float*)d_out;

    const int B = in_sizes[2];           // taskspredids has B elements
    dim3 grid((unsigned)(B / BPB)), block(THREADS);
    trust_scan_kernel<<<grid, block, 0, stream>>>(perf, obsids, predids, obsM, out, B);
}